// WordProbEncoderTD_22497038696524
// MI455X (gfx1250) — compile-verified
//
#include <hip/hip_runtime.h>
#include <hip/hip_bf16.h>

typedef float v2f __attribute__((ext_vector_type(2)));
typedef float v8f __attribute__((ext_vector_type(8)));

#define B_  4
#define L_  256
#define DM_ 256
#define DL_ 64
#define R_  256
// M = B_*L_ = 1024 rows total

// ---------------- Kernel 1: qz = mask ? softmax(x, axis=-1) : 0 ----------------
__global__ void k_softmax(const float* __restrict__ x, const int* __restrict__ mask,
                          float* __restrict__ qz) {
  const int row  = blockIdx.x;          // 0..1023  (z*L + i)
  const int t    = threadIdx.x;         // 0..255
  const int lane = t & 31, wave = t >> 5;
  __shared__ float red[8];

  const float v = x[row * DM_ + t];

  // row max
  float m = v;
  #pragma unroll
  for (int off = 16; off; off >>= 1) m = fmaxf(m, __shfl_xor(m, off, 32));
  if (lane == 0) red[wave] = m;
  __syncthreads();
  float rm = red[0];
  #pragma unroll
  for (int w = 1; w < 8; ++w) rm = fmaxf(rm, red[w]);
  __syncthreads();

  // exp + row sum
  const float e = expf(v - rm);
  float s = e;
  #pragma unroll
  for (int off = 16; off; off >>= 1) s += __shfl_xor(s, off, 32);
  if (lane == 0) red[wave] = s;
  __syncthreads();
  float rs = 0.f;
  #pragma unroll
  for (int w = 0; w < 8; ++w) rs += red[w];

  qz[row * DM_ + t] = (mask[row] != 0) ? (e / rs) : 0.f;
}

// ---------------- Kernel 2: P_s = qz @ V_s  (s = 0,1), fp32 WMMA ----------------
// One 16x16 output tile per wave, K-loop in steps of 4 (V_WMMA_F32_16X16X4_F32).
__global__ void k_gemm_P(const float* __restrict__ qz, const float* __restrict__ V,
                         float* __restrict__ P) {
  const int tid  = threadIdx.x;
  const int lane = tid & 31, wave = tid >> 5;
  const int t    = blockIdx.x * 8 + wave;       // 0..2047 tiles (2 outputs x 64 x 16)
  const int s    = t >> 10;                     // which V / which P
  const int tn   = t & 15;
  const int tm   = (t >> 4) & 63;

  const float* __restrict__ Vs = V + s * (DM_ * R_);
  float* __restrict__       Ps = P + s * (B_ * L_ * R_);

  const int mrow  = tm * 16 + (lane & 15);      // A rows: lanes 0-15 == lanes 16-31
  const int ncol  = tn * 16 + (lane & 15);      // B/D cols
  const int khalf = (lane >> 4) << 1;           // lanes 16-31 hold K+2

  v8f c = {};
  #pragma unroll 8
  for (int k0 = 0; k0 < DM_; k0 += 4) {
    const int kb = k0 + khalf;
    v2f a = *(const v2f*)(qz + mrow * DM_ + kb);     // A[m][kb], A[m][kb+1]
    v2f b;
    b.x = Vs[kb * R_ + ncol];                        // B[kb][n]
    b.y = Vs[(kb + 1) * R_ + ncol];                  // B[kb+1][n]
    c = __builtin_amdgcn_wmma_f32_16x16x4_f32(false, a, false, b, (short)0, c, false, false);
  }

  const int rbase = tm * 16 + ((lane >> 4) << 3);    // lane-half -> M+8
  #pragma unroll
  for (int v = 0; v < 8; ++v)
    Ps[(rbase + v) * R_ + ncol] = c[v];
}

// ---------------- Kernel 3: S_s[z,d] = sum_i P_s[z,i,d];  w_s[d] = sum_c W_s[c,d] ----------------
__global__ void k_sums(const float* __restrict__ P, const float* __restrict__ W,
                       float* __restrict__ S, float* __restrict__ wsum) {
  const int d  = threadIdx.x;
  const int zb = blockIdx.x;
  if (zb < B_) {
    const float* __restrict__ P0 = P;
    const float* __restrict__ P1 = P + B_ * L_ * R_;
    float a0 = 0.f, a1 = 0.f;
    for (int i = 0; i < L_; ++i) {
      a0 += P0[(zb * L_ + i) * R_ + d];
      a1 += P1[(zb * L_ + i) * R_ + d];
    }
    S[zb * R_ + d]            = a0;   // S0
    S[B_ * R_ + zb * R_ + d]  = a1;   // S1
  } else {
    float a0 = 0.f, a1 = 0.f;
    #pragma unroll
    for (int cc = 0; cc < DL_; ++cc) {
      a0 += W[cc * R_ + d];                 // W[0,0,c,d]
      a1 += W[DL_ * R_ + cc * R_ + d];      // W[1,0,c,d]
    }
    wsum[d]      = a0;
    wsum[R_ + d] = a1;
  }
}

// ---------------- Kernel 4: out = x + T0@U0^T + T1@U1^T ----------------
// A-fragments built on the fly: T_s[m,d] = scale[m] * w_s[d] * (S_s[z,d] - P_s[m,d]),
// scale[m] = mask[m] ? 1/64 : 0.  C initialized from x, both terms accumulated in-register.
__global__ void k_gemm_out(const float* __restrict__ x, const float* __restrict__ U,
                           const int* __restrict__ mask,
                           const float* __restrict__ P, const float* __restrict__ S,
                           const float* __restrict__ wsum,
                           float* __restrict__ out) {
  const int tid  = threadIdx.x;
  const int lane = tid & 31, wave = tid >> 5;
  const int t    = blockIdx.x * 8 + wave;   // 0..1023 tiles
  const int tn   = t & 15;
  const int tm   = t >> 4;                  // 0..63
  const int z    = tm >> 4;                 // uniform per tile (16 row-tiles per batch)

  const int mrow  = tm * 16 + (lane & 15);
  const int ncol  = tn * 16 + (lane & 15);
  const int khalf = (lane >> 4) << 1;

  const float scale = (mask[mrow] != 0) ? (1.f / 64.f) : 0.f;

  const float* __restrict__ P0 = P;
  const float* __restrict__ P1 = P + B_ * L_ * R_;
  const float* __restrict__ S0 = S + z * R_;
  const float* __restrict__ S1 = S + B_ * R_ + z * R_;
  const float* __restrict__ w0 = wsum;
  const float* __restrict__ w1 = wsum + R_;
  const float* __restrict__ U0 = U;
  const float* __restrict__ U1 = U + DM_ * R_;

  const int rbase = tm * 16 + ((lane >> 4) << 3);
  v8f c;
  #pragma unroll
  for (int v = 0; v < 8; ++v) c[v] = x[(rbase + v) * DM_ + ncol];   // C := x

  #pragma unroll 4
  for (int k0 = 0; k0 < R_; k0 += 4) {
    const int kb = k0 + khalf;
    v2f p0 = *(const v2f*)(P0 + mrow * R_ + kb);
    v2f p1 = *(const v2f*)(P1 + mrow * R_ + kb);
    v2f a0, a1;
    a0.x = scale * w0[kb]     * (S0[kb]     - p0.x);
    a0.y = scale * w0[kb + 1] * (S0[kb + 1] - p0.y);
    a1.x = scale * w1[kb]     * (S1[kb]     - p1.x);
    a1.y = scale * w1[kb + 1] * (S1[kb + 1] - p1.y);
    v2f b0 = *(const v2f*)(U0 + ncol * R_ + kb);   // B0[k][n] = U0[n,k]
    v2f b1 = *(const v2f*)(U1 + ncol * R_ + kb);
    c = __builtin_amdgcn_wmma_f32_16x16x4_f32(false, a0, false, b0, (short)0, c, false, false);
    c = __builtin_amdgcn_wmma_f32_16x16x4_f32(false, a1, false, b1, (short)0, c, false, false);
  }

  #pragma unroll
  for (int v = 0; v < 8; ++v)
    out[(rbase + v) * DM_ + ncol] = c[v];
}

extern "C" void kernel_launch(void* const* d_in, const int* in_sizes, int n_in,
                              void* d_out, int out_size, void* d_ws, size_t ws_size,
                              hipStream_t stream) {
  (void)in_sizes; (void)n_in; (void)out_size; (void)ws_size;
  const float* x    = (const float*)d_in[0];   // (4,256,256)
  const float* U    = (const float*)d_in[1];   // (2,1,256,256)
  const float* V    = (const float*)d_in[2];   // (2,1,256,256)
  const float* W    = (const float*)d_in[3];   // (2,1,64,256)
  const int*   mask = (const int*)d_in[4];     // (4,256)
  float* out = (float*)d_out;

  float* ws   = (float*)d_ws;
  float* qz   = ws;                    // 262144 floats
  float* P    = ws + 262144;           // 2 * 262144 floats (P0, P1)
  float* S    = ws + 3 * 262144;       // 2 * 1024 floats  (S0, S1)
  float* wsum = S + 2 * B_ * R_;       // 2 * 256 floats   (w0, w1)

  k_softmax <<<B_ * L_, 256, 0, stream>>>(x, mask, qz);
  k_gemm_P  <<<256,     256, 0, stream>>>(qz, V, P);
  k_sums    <<<B_ + 1,  256, 0, stream>>>(P, W, S, wsum);
  k_gemm_out<<<128,     256, 0, stream>>>(x, U, mask, P, S, wsum, out);
}